// GNNEncoder_48421461295706
// MI455X (gfx1250) — compile-verified
//
#include <hip/hip_runtime.h>

// ============================================================================
// GNN encoder for MI455X (gfx1250).
//  - 4x GCN layer: WMMA fp32 GEMM (v_wmma_f32_16x16x4_f32), edge scatter-add
//    via global_atomic_add_f32 (working set is L2-resident: ~180MB < 192MB L2)
//  - pooling via float atomics + bit-pattern atomicMax (values >= 0)
//  - MLP + eval-BatchNorm fused.
// ============================================================================

typedef __attribute__((ext_vector_type(2))) float v2f;
typedef __attribute__((ext_vector_type(8))) float v8f;

#define HD 128
#define BN_EPS 1e-5f

// ---------------- degree / normalization ----------------
__global__ void k_deg_init(float* deg, int n) {
  int i = blockIdx.x * blockDim.x + threadIdx.x;
  if (i < n) deg[i] = 1.0f;  // self loop contributes 1
}

__global__ void k_deg_count(const long long* __restrict__ ei, float* deg, int nE) {
  int e = blockIdx.x * blockDim.x + threadIdx.x;
  if (e < nE) atomicAdd(&deg[(int)ei[(size_t)nE + e]], 1.0f);
}

__global__ void k_dinv(const float* __restrict__ deg, float* dinv, int n) {
  int i = blockIdx.x * blockDim.x + threadIdx.x;
  if (i < n) dinv[i] = rsqrtf(deg[i]);  // deg >= 1 always (self loops)
}

__global__ void k_enorm(const long long* __restrict__ ei, const float* __restrict__ dinv,
                        float* enorm, int nE) {
  int e = blockIdx.x * blockDim.x + threadIdx.x;
  if (e < nE) enorm[e] = dinv[(int)ei[e]] * dinv[(int)ei[(size_t)nE + e]];
}

// ---------------- W transpose (for b64 B-fragment loads) ----------------
__global__ void k_transpose(const float* __restrict__ W, float* __restrict__ Wt) {
  int i = blockIdx.x * blockDim.x + threadIdx.x;  // 128*128 threads
  int k = i >> 7, n = i & 127;
  Wt[n * HD + k] = W[i];
}

// ---------------- WMMA fp32 GEMM:  T[N,128] = Hm[N,128] @ W[128,128] --------
// One 16x16 C tile per wave; 8 waves per block -> 16 rows x 128 cols per block.
// A frag (16x4 f32, 2 VGPRs): lane = M (mod 16), lane>>4 selects K pair.
// B frag (4x16 f32, 2 VGPRs): lane = N (mod 16), lane>>4 selects K pair;
//   loaded as b64 from transposed W.
__global__ void k_gemm_wmma(const float* __restrict__ Hm, const float* __restrict__ Wt,
                            float* __restrict__ T, int nrows) {
  int lane = threadIdx.x & 31;
  int wave = threadIdx.x >> 5;
  int m0 = blockIdx.x << 4;
  int n0 = wave << 4;
  int half = lane >> 4;  // 0 or 1
  int lm = lane & 15;

  int mrow = m0 + lm;
  if (mrow >= nrows) mrow = nrows - 1;  // clamp loads; stores are guarded
  int kk = half << 1;

  const float* arow = Hm + (size_t)mrow * HD;
  const float* brow = Wt + (size_t)(n0 + lm) * HD;

  v8f c = {};
#pragma unroll
  for (int k = 0; k < HD; k += 4) {
    v2f a = *(const v2f*)(arow + k + kk);
    v2f b = *(const v2f*)(brow + k + kk);
    c = __builtin_amdgcn_wmma_f32_16x16x4_f32(false, a, false, b, (short)0, c,
                                              false, false);
  }

  // C layout: VGPR r -> row (m0 + r + half*8), col (n0 + lane&15)
  int rbase = m0 + (half << 3);
  int col = n0 + lm;
#pragma unroll
  for (int r = 0; r < 8; ++r) {
    int row = rbase + r;
    if (row < nrows) T[(size_t)row * HD + col] = c[r];
  }
}

// ---------------- aggregation ----------------
// acc[i] = bias + T[i] * dinv[i]^2   (self-loop term)
__global__ void k_agg_init(const float* __restrict__ T, const float* __restrict__ dinv,
                           const float* __restrict__ bias, float* __restrict__ acc, int n) {
  int t = blockIdx.x * blockDim.x + threadIdx.x;
  int i = t >> 5;
  if (i >= n) return;
  int f = (t & 31) << 2;
  float di = dinv[i];
  float s = di * di;
  float4 v = *(const float4*)(T + (size_t)i * HD + f);
  float4 b = *(const float4*)(bias + f);
  float4 o = make_float4(b.x + v.x * s, b.y + v.y * s, b.z + v.z * s, b.w + v.w * s);
  *(float4*)(acc + (size_t)i * HD + f) = o;
}

// acc[dst] += T[src] * enorm[e]   (32 lanes per edge, float4 per lane)
__global__ void k_agg_edges(const long long* __restrict__ ei, const float* __restrict__ enorm,
                            const float* __restrict__ T, float* __restrict__ acc, int nE) {
  int t = blockIdx.x * blockDim.x + threadIdx.x;
  int e = t >> 5;
  if (e >= nE) return;
  int f = (t & 31) << 2;
  int src = (int)ei[e];
  int dst = (int)ei[(size_t)nE + e];
  float nrm = enorm[e];
  float4 v = *(const float4*)(T + (size_t)src * HD + f);
  float* o = acc + (size_t)dst * HD + f;
  atomicAdd(o + 0, v.x * nrm);
  atomicAdd(o + 1, v.y * nrm);
  atomicAdd(o + 2, v.z * nrm);
  atomicAdd(o + 3, v.w * nrm);
}

// out = relu(acc) (+ res)
__global__ void k_act(const float* __restrict__ acc, const float* __restrict__ res,
                      float* __restrict__ out, int n) {
  int t = blockIdx.x * blockDim.x + threadIdx.x;
  int i = t >> 5;
  if (i >= n) return;
  int f = (t & 31) << 2;
  size_t off = (size_t)i * HD + f;
  float4 a = *(const float4*)(acc + off);
  float4 o = make_float4(fmaxf(a.x, 0.f), fmaxf(a.y, 0.f), fmaxf(a.z, 0.f), fmaxf(a.w, 0.f));
  if (res) {
    float4 r = *(const float4*)(res + off);
    o.x += r.x; o.y += r.y; o.z += r.z; o.w += r.w;
  }
  *(float4*)(out + off) = o;
}

// ---------------- pooling ----------------
__global__ void k_zero(float* p, int n) {
  int i = blockIdx.x * blockDim.x + threadIdx.x;
  if (i < n) p[i] = 0.0f;
}

__global__ void k_pool(const float* __restrict__ X, const long long* __restrict__ batch,
                       float* __restrict__ cnt, float* __restrict__ psum,
                       unsigned int* __restrict__ pmaxb, int n) {
  int t = blockIdx.x * blockDim.x + threadIdx.x;
  int i = t >> 5;
  if (i >= n) return;
  int f = (t & 31) << 2;
  int g = (int)batch[i];
  float4 v = *(const float4*)(X + (size_t)i * HD + f);
  float* ps = psum + (size_t)g * HD + f;
  atomicAdd(ps + 0, v.x);
  atomicAdd(ps + 1, v.y);
  atomicAdd(ps + 2, v.z);
  atomicAdd(ps + 3, v.w);
  // all values >= 0 (relu chain), so float bit pattern is monotone in uint
  unsigned int* pm = pmaxb + (size_t)g * HD + f;
  atomicMax(pm + 0, __float_as_uint(v.x));
  atomicMax(pm + 1, __float_as_uint(v.y));
  atomicMax(pm + 2, __float_as_uint(v.z));
  atomicMax(pm + 3, __float_as_uint(v.w));
  if ((t & 31) == 0) atomicAdd(cnt + g, 1.0f);
}

// ---------------- MLP + eval BatchNorm ----------------
__global__ void k_mlp1(const float* __restrict__ psum, const float* __restrict__ pmax,
                       const float* __restrict__ cnt, const float* __restrict__ P1,
                       const float* __restrict__ pb1, const float* __restrict__ g1,
                       const float* __restrict__ be1, const float* __restrict__ rm1,
                       const float* __restrict__ rv1, float* __restrict__ hmid, int G) {
  int t = blockIdx.x * blockDim.x + threadIdx.x;  // G * 256
  int g = t >> 8;
  int o = t & 255;
  if (g >= G) return;
  float cinv = 1.0f / fmaxf(cnt[g], 1.0f);
  const float* ps = psum + (size_t)g * HD;
  const float* pm = pmax + (size_t)g * HD;
  float s = pb1[o];
  for (int j = 0; j < HD; ++j) s += ps[j] * cinv * P1[j * 256 + o];          // mean
  for (int j = 0; j < HD; ++j) s += pm[j] * P1[(HD + j) * 256 + o];          // max
  for (int j = 0; j < HD; ++j) s += ps[j] * P1[(2 * HD + j) * 256 + o];      // sum
  s = (s - rm1[o]) * rsqrtf(rv1[o] + BN_EPS) * g1[o] + be1[o];
  hmid[(size_t)g * 256 + o] = fmaxf(s, 0.0f);
}

__global__ void k_mlp2(const float* __restrict__ hmid, const float* __restrict__ P2,
                       const float* __restrict__ pb2, const float* __restrict__ g2,
                       const float* __restrict__ be2, const float* __restrict__ rm2,
                       const float* __restrict__ rv2, float* __restrict__ z, int G) {
  int t = blockIdx.x * blockDim.x + threadIdx.x;  // G * 128
  int g = t >> 7;
  int o = t & 127;
  if (g >= G) return;
  const float* h = hmid + (size_t)g * 256;
  float s = pb2[o];
  for (int j = 0; j < 256; ++j) s += h[j] * P2[j * HD + o];
  z[(size_t)g * HD + o] = (s - rm2[o]) * rsqrtf(rv2[o] + BN_EPS) * g2[o] + be2[o];
}

// ============================================================================
static inline size_t align4(size_t x) { return (x + 3) & ~(size_t)3; }

extern "C" void kernel_launch(void* const* d_in, const int* in_sizes, int n_in,
                              void* d_out, int out_size, void* d_ws, size_t ws_size,
                              hipStream_t stream) {
  const float* x = (const float*)d_in[0];
  const long long* ei = (const long long*)d_in[1];
  const long long* batch = (const long long*)d_in[2];
  const float* Wl[4] = {(const float*)d_in[3], (const float*)d_in[5],
                        (const float*)d_in[7], (const float*)d_in[9]};
  const float* bl[4] = {(const float*)d_in[4], (const float*)d_in[6],
                        (const float*)d_in[8], (const float*)d_in[10]};
  const float* P1 = (const float*)d_in[11];
  const float* pb1 = (const float*)d_in[12];
  const float* g1 = (const float*)d_in[13];
  const float* be1 = (const float*)d_in[14];
  const float* rm1 = (const float*)d_in[15];
  const float* rv1 = (const float*)d_in[16];
  const float* P2 = (const float*)d_in[17];
  const float* pb2 = (const float*)d_in[18];
  const float* g2 = (const float*)d_in[19];
  const float* be2 = (const float*)d_in[20];
  const float* rm2 = (const float*)d_in[21];
  const float* rv2 = (const float*)d_in[22];

  const int N = in_sizes[2];       // batch is [N]
  const int E = in_sizes[1] / 2;   // edge_index is [2, E]
  const int G = out_size / HD;     // output is [G, 128]

  // ---- workspace layout (floats) ----
  float* ws = (float*)d_ws;
  size_t off = 0;
  float* deg = ws + off;   off += align4((size_t)N);
  float* dinv = ws + off;  off += align4((size_t)N);
  float* enorm = ws + off; off += align4((size_t)E);
  float* bufP = ws + off;  off += (size_t)N * HD;
  float* bufQ = ws + off;  off += (size_t)N * HD;
  float* bufR = ws + off;  off += (size_t)N * HD;   // aggregation accumulator
  float* Wt = ws + off;    off += (size_t)HD * HD;
  float* cnt = ws + off;   off += align4((size_t)G);
  float* psum = ws + off;  off += (size_t)G * HD;
  float* pmax = ws + off;  off += (size_t)G * HD;
  float* hmid = ws + off;  off += (size_t)G * 256;
  (void)ws_size;

  const int TB = 256;
  int bN = (N + TB - 1) / TB;
  int bE = (E + TB - 1) / TB;
  int bN32 = (N * 32 + TB - 1) / TB;   // node x feature-quad threads
  int bE32 = ((size_t)E * 32 + TB - 1) / TB;

  // ---- normalization precompute ----
  k_deg_init<<<bN, TB, 0, stream>>>(deg, N);
  k_deg_count<<<bE, TB, 0, stream>>>(ei, deg, E);
  k_dinv<<<bN, TB, 0, stream>>>(deg, dinv, N);
  k_enorm<<<bE, TB, 0, stream>>>(ei, dinv, enorm, E);

  // ---- 4 GCN layers, ping-pong P/Q with accumulator R ----
  int gemmBlocks = (N + 15) / 16;
  const float* in = x;
  float* tb = bufP;
  for (int l = 0; l < 4; ++l) {
    k_transpose<<<(HD * HD) / TB, TB, 0, stream>>>(Wl[l], Wt);
    k_gemm_wmma<<<gemmBlocks, TB, 0, stream>>>(in, Wt, tb, N);
    k_agg_init<<<bN32, TB, 0, stream>>>(tb, dinv, bl[l], bufR, N);
    k_agg_edges<<<bE32, TB, 0, stream>>>(ei, enorm, tb, bufR, E);
    k_act<<<bN32, TB, 0, stream>>>(bufR, (l == 0) ? nullptr : in, tb, N);
    in = tb;                       // layer output lives in tb
    tb = (tb == bufP) ? bufQ : bufP;
  }
  const float* x4 = in;

  // ---- pooling (cnt, psum, pmax are contiguous -> one zero pass) ----
  int zn = align4((size_t)G) + 2 * G * HD;
  k_zero<<<(zn + TB - 1) / TB, TB, 0, stream>>>(cnt, zn);
  k_pool<<<bN32, TB, 0, stream>>>(x4, batch, cnt, psum, (unsigned int*)pmax, N);

  // ---- MLP head ----
  k_mlp1<<<(G * 256) / TB, TB, 0, stream>>>(psum, pmax, cnt, P1, pb1, g1, be1, rm1, rv1,
                                            hmid, G);
  k_mlp2<<<(G * HD) / TB, TB, 0, stream>>>(hmid, P2, pb2, g2, be2, rm2, rv2,
                                           (float*)d_out, G);
}